// MessagePassingLayer_59846074303161
// MI455X (gfx1250) — compile-verified
//
#include <hip/hip_runtime.h>
#include <hip/hip_bf16.h>

// MessagePassingLayer for MI455X (gfx1250, wave32, WMMA).
// Stage 1: per-edge  msg = relu(x[src] @ W_msg + b) * w   -> atomic scatter-add into agg[dst], deg[dst]+=1
//          Persistent waves; W_msg held entirely in VGPRs (128 regs/lane), zero LDS traffic,
//          wave-uniform fast path for full tiles (no per-element EXEC juggling in the hot loop).
// Stage 2: per-node  h   = relu([x || agg/max(deg,1)] @ W_upd + b), then row L2-normalize.
// Both GEMMs use V_WMMA_F32_16X16X4_F32 (fp32 matrix path, matches fp32 reference numerics).

typedef float v2f __attribute__((ext_vector_type(2)));
typedef float v8f __attribute__((ext_vector_type(8)));

#define DIN  64
#define DOUT 64
#define KUPD 128  // DIN + DOUT

// ---------------------------------------------------------------------------
__global__ void zero_f32_kernel(float* __restrict__ p, long n) {
    long i = (long)blockIdx.x * blockDim.x + threadIdx.x;
    long stride = (long)gridDim.x * blockDim.x;
    for (; i < n; i += stride) p[i] = 0.0f;
}

// ---------------------------------------------------------------------------
// Edge kernel: persistent, 8 waves/block, each wave grid-strides over 16-edge tiles.
// B operand (W_msg) is loaded once per wave into registers: barr[16][4] v2f = 128 VGPRs.
__global__ void __launch_bounds__(256)
edge_msg_kernel(const float* __restrict__ x,
                const int*   __restrict__ src_idx,
                const int*   __restrict__ dst_idx,
                const float* __restrict__ ew,
                const float* __restrict__ W_msg,   // [64,64] row-major (k,n)
                const float* __restrict__ b_msg,   // [64]
                float* __restrict__ agg,           // [N,64]
                float* __restrict__ deg,           // [N]
                int E, int ntiles)
{
    const int lane = threadIdx.x & 31;
    const int m  = lane & 15;                      // A row / B,C column index
    const int kh = lane >> 4;                      // K half select (0 or 1)

    // ---- Prologue: B matrix resident in VGPRs for the lifetime of the wave ----
    v2f barr[16][4];
    #pragma unroll
    for (int kk = 0; kk < 16; ++kk) {
        const int k0 = kk * 4 + kh * 2;            // this lane's K pair {k0, k0+1}
        const float* wr0 = W_msg + (k0    ) * DOUT;
        const float* wr1 = W_msg + (k0 + 1) * DOUT;
        #pragma unroll
        for (int t = 0; t < 4; ++t) {
            barr[kk][t].x = wr0[t * 16 + m];
            barr[kk][t].y = wr1[t * 16 + m];
        }
    }
    const float bb0 = b_msg[ 0 + m];
    const float bb1 = b_msg[16 + m];
    const float bb2 = b_msg[32 + m];
    const float bb3 = b_msg[48 + m];

    const int waves_per_blk = blockDim.x >> 5;
    const int gwave   = blockIdx.x * waves_per_blk + (threadIdx.x >> 5);
    const int wstride = gridDim.x * waves_per_blk;

    // ---- Steady state: 64 WMMA + 16 A-loads + metadata + scatter per 16-edge tile ----
    for (int tile = gwave; tile < ntiles; tile += wstride) {
        const int e0 = tile * 16;
        const bool full = (e0 + 16 <= E);          // wave-uniform

        if (full) {
            // ------------------ fast path: no per-element guards ------------------
            const float* __restrict__ xrow = x + (long)src_idx[e0 + m] * DIN;

            // Metadata first so these loads pipeline under the WMMA A-load stream.
            int   dsts[8];
            float ws[8];
            #pragma unroll
            for (int r = 0; r < 8; ++r) {
                const int e = e0 + r + 8 * kh;     // this lane's C/D row r -> edge e
                dsts[r] = dst_idx[e];
                ws[r]   = ew[e];
            }

            v8f acc0 = {}, acc1 = {}, acc2 = {}, acc3 = {};
            #pragma unroll
            for (int kk = 0; kk < 16; ++kk) {      // K = 64 in steps of 4
                const int k0 = kk * 4 + kh * 2;
                v2f a = *(const v2f*)(xrow + k0);  // global_load_b64
                acc0 = __builtin_amdgcn_wmma_f32_16x16x4_f32(false, a, false, barr[kk][0], (short)0, acc0, false, false);
                acc1 = __builtin_amdgcn_wmma_f32_16x16x4_f32(false, a, false, barr[kk][1], (short)0, acc1, false, false);
                acc2 = __builtin_amdgcn_wmma_f32_16x16x4_f32(false, a, false, barr[kk][2], (short)0, acc2, false, false);
                acc3 = __builtin_amdgcn_wmma_f32_16x16x4_f32(false, a, false, barr[kk][3], (short)0, acc3, false, false);
            }

            #pragma unroll
            for (int r = 0; r < 8; ++r) {
                const float w    = ws[r];
                const long  base = (long)dsts[r] * DOUT;
                atomicAdd(&agg[base +  0 + m], fmaxf(acc0[r] + bb0, 0.0f) * w);
                atomicAdd(&agg[base + 16 + m], fmaxf(acc1[r] + bb1, 0.0f) * w);
                atomicAdd(&agg[base + 32 + m], fmaxf(acc2[r] + bb2, 0.0f) * w);
                atomicAdd(&agg[base + 48 + m], fmaxf(acc3[r] + bb3, 0.0f) * w);
            }
            if (kh == 0) atomicAdd(&deg[dst_idx[e0 + m]], 1.0f);
        } else {
            // ------------------ tail path: fully guarded (cold) ------------------
            int em = e0 + m; if (em >= E) em = E - 1;
            const float* __restrict__ xrow = x + (long)src_idx[em] * DIN;

            v8f acc0 = {}, acc1 = {}, acc2 = {}, acc3 = {};
            #pragma unroll
            for (int kk = 0; kk < 16; ++kk) {
                const int k0 = kk * 4 + kh * 2;
                v2f a = *(const v2f*)(xrow + k0);
                acc0 = __builtin_amdgcn_wmma_f32_16x16x4_f32(false, a, false, barr[kk][0], (short)0, acc0, false, false);
                acc1 = __builtin_amdgcn_wmma_f32_16x16x4_f32(false, a, false, barr[kk][1], (short)0, acc1, false, false);
                acc2 = __builtin_amdgcn_wmma_f32_16x16x4_f32(false, a, false, barr[kk][2], (short)0, acc2, false, false);
                acc3 = __builtin_amdgcn_wmma_f32_16x16x4_f32(false, a, false, barr[kk][3], (short)0, acc3, false, false);
            }

            #pragma unroll
            for (int r = 0; r < 8; ++r) {
                const int e = e0 + r + 8 * kh;
                if (e < E) {
                    const float w    = ew[e];
                    const long  base = (long)dst_idx[e] * DOUT;
                    atomicAdd(&agg[base +  0 + m], fmaxf(acc0[r] + bb0, 0.0f) * w);
                    atomicAdd(&agg[base + 16 + m], fmaxf(acc1[r] + bb1, 0.0f) * w);
                    atomicAdd(&agg[base + 32 + m], fmaxf(acc2[r] + bb2, 0.0f) * w);
                    atomicAdd(&agg[base + 48 + m], fmaxf(acc3[r] + bb3, 0.0f) * w);
                }
            }
            if (kh == 0 && e0 + m < E) atomicAdd(&deg[dst_idx[e0 + m]], 1.0f);
        }
    }
}

// ---------------------------------------------------------------------------
// Node kernel: 4 waves/block, each wave owns a tile of 16 nodes. B (W_upd, K=128)
// is too large for registers -> staged in LDS.
__global__ void __launch_bounds__(128)
node_update_kernel(const float* __restrict__ x,
                   const float* __restrict__ agg,
                   const float* __restrict__ deg,
                   const float* __restrict__ W_upd,   // [128,64] row-major (k,n)
                   const float* __restrict__ b_upd,   // [64]
                   float* __restrict__ out,           // [N,64]
                   int N)
{
    __shared__ float Wlds[KUPD * DOUT];                // 32 KB
    __shared__ float hs[4][16 * DOUT];                 // 16 KB (per-wave staging)
    for (int i = threadIdx.x; i < KUPD * DOUT; i += blockDim.x) Wlds[i] = W_upd[i];
    __syncthreads();

    const int lane = threadIdx.x & 31;
    const int wave = threadIdx.x >> 5;
    const int tile = blockIdx.x * 4 + wave;
    const int n0   = tile * 16;
    if (n0 >= N) return;                               // wave-uniform exit

    const int m  = lane & 15;
    const int kh = lane >> 4;

    int node = n0 + m; if (node >= N) node = N - 1;    // clamp tail (stores guarded)
    const float* __restrict__ xrow = x   + (long)node * DIN;
    const float* __restrict__ arow = agg + (long)node * DOUT;
    const float  rdeg = 1.0f / fmaxf(deg[node], 1.0f); // mean-agg denominator, folded into A

    v8f acc0 = {}, acc1 = {}, acc2 = {}, acc3 = {};

    // K = 0..63 : x half of the concat
    #pragma unroll
    for (int kk = 0; kk < 16; ++kk) {
        const int k0 = kk * 4 + kh * 2;
        v2f a = *(const v2f*)(xrow + k0);
        const float* wr0 = &Wlds[(k0    ) * DOUT];
        const float* wr1 = &Wlds[(k0 + 1) * DOUT];
        v2f b0 = { wr0[ 0 + m], wr1[ 0 + m] };
        v2f b1 = { wr0[16 + m], wr1[16 + m] };
        v2f b2 = { wr0[32 + m], wr1[32 + m] };
        v2f b3 = { wr0[48 + m], wr1[48 + m] };
        acc0 = __builtin_amdgcn_wmma_f32_16x16x4_f32(false, a, false, b0, (short)0, acc0, false, false);
        acc1 = __builtin_amdgcn_wmma_f32_16x16x4_f32(false, a, false, b1, (short)0, acc1, false, false);
        acc2 = __builtin_amdgcn_wmma_f32_16x16x4_f32(false, a, false, b2, (short)0, acc2, false, false);
        acc3 = __builtin_amdgcn_wmma_f32_16x16x4_f32(false, a, false, b3, (short)0, acc3, false, false);
    }
    // K = 64..127 : agg half, scaled by 1/max(deg,1)
    #pragma unroll
    for (int kk = 0; kk < 16; ++kk) {
        const int k0 = kk * 4 + kh * 2;
        v2f t = *(const v2f*)(arow + k0);
        v2f a; a.x = t.x * rdeg; a.y = t.y * rdeg;
        const float* wr0 = &Wlds[(64 + k0    ) * DOUT];
        const float* wr1 = &Wlds[(64 + k0 + 1) * DOUT];
        v2f b0 = { wr0[ 0 + m], wr1[ 0 + m] };
        v2f b1 = { wr0[16 + m], wr1[16 + m] };
        v2f b2 = { wr0[32 + m], wr1[32 + m] };
        v2f b3 = { wr0[48 + m], wr1[48 + m] };
        acc0 = __builtin_amdgcn_wmma_f32_16x16x4_f32(false, a, false, b0, (short)0, acc0, false, false);
        acc1 = __builtin_amdgcn_wmma_f32_16x16x4_f32(false, a, false, b1, (short)0, acc1, false, false);
        acc2 = __builtin_amdgcn_wmma_f32_16x16x4_f32(false, a, false, b2, (short)0, acc2, false, false);
        acc3 = __builtin_amdgcn_wmma_f32_16x16x4_f32(false, a, false, b3, (short)0, acc3, false, false);
    }

    // Bias + ReLU, stage h tile in this wave's LDS slab (rows r+8*kh, col 16*t+m)
    const float bb0 = b_upd[ 0 + m];
    const float bb1 = b_upd[16 + m];
    const float bb2 = b_upd[32 + m];
    const float bb3 = b_upd[48 + m];
    float* __restrict__ h = hs[wave];
    #pragma unroll
    for (int r = 0; r < 8; ++r) {
        const int row = r + 8 * kh;
        h[row * DOUT +  0 + m] = fmaxf(acc0[r] + bb0, 0.0f);
        h[row * DOUT + 16 + m] = fmaxf(acc1[r] + bb1, 0.0f);
        h[row * DOUT + 32 + m] = fmaxf(acc2[r] + bb2, 0.0f);
        h[row * DOUT + 48 + m] = fmaxf(acc3[r] + bb3, 0.0f);
    }
    // Same-wave LDS RAW across lanes: drain DS pipe (in-order per wave, but be explicit).
    asm volatile("s_wait_dscnt 0" ::: "memory");

    // Row L2 norm: 2 lanes per row, each sums 32 squares, combine via wave32 shfl_xor.
    const int row  = lane >> 1;         // 0..15
    const int half = lane & 1;          // 0/1 -> columns [32*half, 32*half+32)
    float ssq = 0.0f;
    #pragma unroll
    for (int j = 0; j < 32; ++j) {
        float v = h[row * DOUT + half * 32 + j];
        ssq += v * v;
    }
    ssq += __shfl_xor(ssq, 1, 32);
    const float inv = 1.0f / fmaxf(sqrtf(ssq), 1e-12f);

    const int onode = n0 + row;
    if (onode < N) {
        float* __restrict__ orow = out + (long)onode * DOUT + half * 32;
        #pragma unroll
        for (int j = 0; j < 32; ++j)
            orow[j] = h[row * DOUT + half * 32 + j] * inv;
    }
}

// ---------------------------------------------------------------------------
extern "C" void kernel_launch(void* const* d_in, const int* in_sizes, int n_in,
                              void* d_out, int out_size, void* d_ws, size_t ws_size,
                              hipStream_t stream) {
    const float* x       = (const float*)d_in[0];   // [N,64]
    const int*   eidx    = (const int*)  d_in[1];   // [2,E]
    const float* ew      = (const float*)d_in[2];   // [E]
    const float* W_msg   = (const float*)d_in[3];   // [64,64]
    const float* b_msg   = (const float*)d_in[4];   // [64]
    const float* W_upd   = (const float*)d_in[5];   // [128,64]
    const float* b_upd   = (const float*)d_in[6];   // [64]
    float*       out     = (float*)d_out;

    const int N = in_sizes[0] / DIN;
    const int E = in_sizes[2];
    const int* src_idx = eidx;
    const int* dst_idx = eidx + E;

    float* agg = (float*)d_ws;                      // [N,64]
    float* deg = agg + (size_t)N * DOUT;            // [N]

    // Zero the aggregation scratch (contiguous agg||deg region)
    const long nz = (long)N * DOUT + N;
    zero_f32_kernel<<<2048, 256, 0, stream>>>(agg, nz);

    // Edge stage: persistent waves, 16 edges per tile, 8 waves per block.
    const int etiles  = (E + 15) / 16;
    int eblocks = (etiles + 7) / 8;
    if (eblocks > 640) eblocks = 640;               // ~5120 waves, ~15 tiles/wave
    edge_msg_kernel<<<eblocks, 256, 0, stream>>>(x, src_idx, dst_idx, ew,
                                                 W_msg, b_msg, agg, deg, E, etiles);

    // Node stage: 16 nodes per wave, 4 waves per block
    const int ntiles  = (N + 15) / 16;
    const int nblocks = (ntiles + 3) / 4;
    node_update_kernel<<<nblocks, 128, 0, stream>>>(x, agg, deg, W_upd, b_upd, out, N);
}